// ADMMSolverBlock_16209206575821
// MI455X (gfx1250) — compile-verified
//
#include <hip/hip_runtime.h>
#include <math.h>

// ---------------------------------------------------------------------------
// ADMM TV solver, spectral (DCT-II) solve of (I + rho*(Dx^T Dx + Dy^T Dy))^-1
// H = W = 64, N = 4096, BATCH = 64, channels = 2, T = 4
// State layout in ws (float): [c][i][j][b], b fastest (64 contiguous floats).
// ---------------------------------------------------------------------------

#define Hh   64
#define Ww   64
#define Bb   64
#define NE   (2 * Hh * Ww * Bb)        // 524288 elements per [2][H][W][B] tensor
#define OPL  (2 * Hh * Ww * Bb * 2)    // Z/Beta: [2 op][2 c][H][W][B] = 2*NE
#define RHO_F  0.1f
#define ETA_F  0.1f
#define TH_F   0.1f                    // lamb/rho = 0.01/0.1

typedef float v2f __attribute__((ext_vector_type(2)));
typedef float v8f __attribute__((ext_vector_type(8)));

// ---------------------------------------------------------------------------
// Table setup: orthonormal DCT-II matrix C, its transpose, and the spectral
// inverse denominators 1/(1 + rho*(lam_k + lam_l)).
// ---------------------------------------------------------------------------
__global__ void build_tables(float* __restrict__ Cmat, float* __restrict__ Ct,
                             float* __restrict__ invDen) {
  int e = blockIdx.x * blockDim.x + threadIdx.x;   // 0..4095
  if (e >= Hh * Hh) return;
  int k = e >> 6, i = e & 63;
  const double PI = 3.14159265358979323846;
  double s = (k == 0) ? sqrt(1.0 / 64.0) : sqrt(2.0 / 64.0);
  double v = s * cos(PI * (double)k * (2.0 * i + 1.0) / 128.0);
  Cmat[k * 64 + i] = (float)v;
  Ct[i * 64 + k]   = (float)v;
  double lamk = 2.0 - 2.0 * cos(PI * (double)k / 64.0);
  double lami = 2.0 - 2.0 * cos(PI * (double)i / 64.0);
  invDen[k * 64 + i] = (float)(1.0 / (1.0 + 0.1 * (lamk + lami)));
}

__global__ void zero4_kernel(float4* __restrict__ p, int n4) {
  int i = blockIdx.x * blockDim.x + threadIdx.x;
  if (i < n4) p[i] = make_float4(0.f, 0.f, 0.f, 0.f);
}

// Fc[c][i][j][b] = F[b][c][63-i][j]   (flip along H)
__global__ void init_fc(const float* __restrict__ F, float* __restrict__ Fc) {
  int g = blockIdx.x * blockDim.x + threadIdx.x;
  if (g >= NE) return;
  int b = g & 63, t = g >> 6;
  int j = t & 63; t >>= 6;
  int i = t & 63; int c = t >> 6;
  Fc[g] = F[(((b * 2 + c) * 64) + (63 - i)) * 64 + j];
}

// R = Fc + rho * ( Dx^T wx + Dy^T wy ),  w = Z - Beta
__global__ void compute_rhs(const float* __restrict__ Fc,
                            const float* __restrict__ Z,
                            const float* __restrict__ Beta,
                            float* __restrict__ R) {
  int g = blockIdx.x * blockDim.x + threadIdx.x;
  if (g >= NE) return;
  int t = g >> 6;
  int j = t & 63; t >>= 6;
  int i = t & 63;
  const int OY = Hh * Ww * Bb * 2;   // offset of op=1 plane (524288)
  float dxT = 0.f, dyT = 0.f;
  if (j > 0)  dxT += Z[g - 64]        - Beta[g - 64];
  if (j < 63) dxT -= Z[g]             - Beta[g];
  if (i > 0)  dyT += Z[g + OY - 4096] - Beta[g + OY - 4096];
  if (i < 63) dyT -= Z[g + OY]        - Beta[g + OY];
  R[g] = Fc[g] + RHO_F * (dxT + dyT);
}

// ---------------------------------------------------------------------------
// WMMA f32 GEMM (fully constant-strided): per slab s,
//   Y[m][n] = sum_k A[m][k] * X[k][n]          (m,k in 0..63, n in 0..NCOLS-1)
// A: 64x64 row-major.  X/Y slab: row stride ROWS, cols contiguous.
// One wave per 16x16 tile; K=64 -> 16 chained V_WMMA_F32_16X16X4_F32.
// Optional epilogue scale (eigen denominators): Y *= scale[(s&63)*64 + m].
// Grid must supply exactly nSlabs * 4 * (NCOLS/16) waves (EXEC all-1s).
// ---------------------------------------------------------------------------
template <int ROWS, int NCOLS, int SLABSTR, bool HAS_SCALE>
__global__ void dct_gemm(const float* __restrict__ A,
                         const float* __restrict__ X,
                         float* __restrict__ Y,
                         const float* __restrict__ scale) {
  constexpr int NTC = NCOLS >> 4;       // tile columns per slab
  constexpr int TPS = 4 * NTC;          // tiles per slab (power of two)
  int waveId = blockIdx.x * (blockDim.x >> 5) + (threadIdx.x >> 5);
  int lane = threadIdx.x & 31;
  int s  = waveId / TPS;                // shift (TPS = 1024 or 16)
  int t  = waveId & (TPS - 1);
  int m0 = (t / NTC) << 4;
  int n0 = (t & (NTC - 1)) << 4;

  int half = lane >> 4;                 // 0: lanes 0-15, 1: lanes 16-31
  int ml = lane & 15;

  // Per-lane bases; all loop offsets below are compile-time constants.
  const float* Ap = A + (m0 + ml) * 64 + 2 * half;
  const float* Xp = X + (size_t)s * SLABSTR + (size_t)(2 * half) * ROWS + n0 + ml;

  v8f acc = {};
#pragma unroll
  for (int kt = 0; kt < 16; ++kt) {
    v2f a;                              // A[m][k0], A[m][k0+1]
    a.x = Ap[kt * 4];
    a.y = Ap[kt * 4 + 1];
    v2f b;                              // X[k0][n], X[k0+1][n]
    b.x = Xp[kt * 4 * ROWS];
    b.y = Xp[kt * 4 * ROWS + ROWS];
    acc = __builtin_amdgcn_wmma_f32_16x16x4_f32(
        false, a, false, b, (short)0, acc, false, false);
  }

  float* Yp = Y + (size_t)s * SLABSTR + (size_t)(m0 + 8 * half) * ROWS + n0 + ml;
  const float* sp = HAS_SCALE ? (scale + ((s & 63) << 6) + m0 + 8 * half)
                              : nullptr;
#pragma unroll
  for (int r = 0; r < 8; ++r) {         // D: row m0 + r + 8*half, col ml
    float v = acc[r];
    if constexpr (HAS_SCALE) v *= sp[r];
    Yp[r * ROWS] = v;
  }
}

// DQ = D*Q; C = DQ + Beta; Z = shrink(C); Beta += eta*(DQ - Z)
__global__ void update_z_beta(const float* __restrict__ Q,
                              float* __restrict__ Z,
                              float* __restrict__ Beta) {
  int g = blockIdx.x * blockDim.x + threadIdx.x;
  if (g >= NE) return;
  int t = g >> 6;
  int j = t & 63; t >>= 6;
  int i = t & 63;
  const int OY = Hh * Ww * Bb * 2;
  float q = Q[g];
  float dqx = (j < 63) ? (Q[g + 64] - q) : 0.f;
  float dqy = (i < 63) ? (Q[g + 4096] - q) : 0.f;

  float cx = dqx + Beta[g];
  float zx = (fabsf(cx) >= TH_F) ? (cx - copysignf(TH_F, cx)) : 0.f;
  Z[g] = zx;
  Beta[g] += ETA_F * (dqx - zx);

  float cy = dqy + Beta[g + OY];
  float zy = (fabsf(cy) >= TH_F) ? (cy - copysignf(TH_F, cy)) : 0.f;
  Z[g + OY] = zy;
  Beta[g + OY] += ETA_F * (dqy - zy);
}

// out[b][c][i][j] = Q[c][63-i][j][b]   (un-flip H, transpose batch out)
__global__ void write_out(const float* __restrict__ Q, float* __restrict__ out) {
  int g = blockIdx.x * blockDim.x + threadIdx.x;
  if (g >= NE) return;
  int j = g & 63, t = g >> 6;
  int i = t & 63; t >>= 6;
  int c = t & 1;  int b = t >> 1;
  out[g] = Q[(((c * 64) + (63 - i)) * 64 + j) * 64 + b];
}

// ---------------------------------------------------------------------------
extern "C" void kernel_launch(void* const* d_in, const int* in_sizes, int n_in,
                              void* d_out, int out_size, void* d_ws, size_t ws_size,
                              hipStream_t stream) {
  (void)in_sizes; (void)n_in; (void)out_size; (void)ws_size;
  const float* F = (const float*)d_in[0];   // [64,2,64,64]; image & D unused
  float* out = (float*)d_out;

  float* w      = (float*)d_ws;
  float* Cmat   = w;                         //  4096
  float* Ctm    = w + 4096;                  //  4096
  float* invDen = w + 8192;                  //  4096
  float* bufA   = w + 12288;                 //  NE  (Fc)
  float* bufB   = bufA + NE;                 //  NE  (Q)
  float* Rb     = bufB + NE;                 //  NE  (rhs / spectral temp)
  float* T1     = Rb + NE;                   //  NE
  float* Z      = T1 + NE;                   //  2*NE
  float* Beta   = Z + OPL;                   //  2*NE   (~16.1 MiB total)

  build_tables<<<16, 256, 0, stream>>>(Cmat, Ctm, invDen);
  zero4_kernel<<<(2 * OPL / 4 + 255) / 256, 256, 0, stream>>>((float4*)Z,
                                                              2 * OPL / 4);
  init_fc<<<NE / 256, 256, 0, stream>>>(F, bufA);

  float* fc = bufA;
  float* qb = bufB;
  constexpr int SLAB = Hh * Ww * Bb;         // 262144 (per-channel slab)
  for (int it = 0; it < 4; ++it) {
    compute_rhs<<<NE / 256, 256, 0, stream>>>(fc, Z, Beta, Rb);
    // spectral solve:
    //   T1 = C *_H R ; Rb = diag * (C *_W T1) ; T1 = Ct *_H Rb ; qb = Ct *_W T1
    // H-passes: 2 slabs * 1024 tiles = 2048 waves -> 512 blocks of 4 waves
    // W-passes: 128 slabs * 16 tiles = 2048 waves -> 512 blocks of 4 waves
    dct_gemm<4096, 4096, SLAB, false><<<512, 128, 0, stream>>>(Cmat, Rb, T1, nullptr);
    dct_gemm<64,   64,   4096, true ><<<512, 128, 0, stream>>>(Cmat, T1, Rb, invDen);
    dct_gemm<4096, 4096, SLAB, false><<<512, 128, 0, stream>>>(Ctm, Rb, T1, nullptr);
    dct_gemm<64,   64,   4096, false><<<512, 128, 0, stream>>>(Ctm, T1, qb, nullptr);
    update_z_beta<<<NE / 256, 256, 0, stream>>>(qb, Z, Beta);
    float* tmp = fc; fc = qb; qb = tmp;      // Fc <- Q for next iteration
  }
  write_out<<<NE / 256, 256, 0, stream>>>(fc, out);
}